// TokenWiseLayerPool_39994735460421
// MI455X (gfx1250) — compile-verified
//
#include <hip/hip_runtime.h>
#include <cmath>
#include <cstdint>

#define L_    32
#define B_    4
#define T_    1024
#define H_    1280
#define HID_  128
#define NTOK  (B_ * T_)
#define BTH   ((size_t)B_ * T_ * H_)
#define NTHR  320   // 10 waves; one float4 of H per thread

typedef __attribute__((ext_vector_type(2))) float v2f;
typedef __attribute__((ext_vector_type(4))) float v4f;
typedef __attribute__((ext_vector_type(8))) float v8f;
typedef __attribute__((ext_vector_type(4))) int   v4i;

#if __has_builtin(__builtin_amdgcn_global_load_async_to_lds_b128)
#define HAVE_ASYNC 1
#pragma message("CDNA5 probe: global_load_async_to_lds_b128 builtin = YES")
#else
#define HAVE_ASYNC 0
#pragma message("CDNA5 probe: global_load_async_to_lds_b128 builtin = NO (sync fallback)")
#endif

#if __has_builtin(__builtin_amdgcn_s_wait_asynccnt)
#define HAVE_WAITASYNC 1
#else
#define HAVE_WAITASYNC 0
#endif

#if __has_builtin(__builtin_amdgcn_wmma_f32_16x16x4_f32)
#define HAVE_WMMA_F32 1
#pragma message("CDNA5 probe: wmma_f32_16x16x4_f32 builtin = YES")
#else
#define HAVE_WMMA_F32 0
#pragma message("CDNA5 probe: wmma_f32_16x16x4_f32 builtin = NO (VALU fallback)")
#endif

#if HAVE_ASYNC
// Param 1 is a global (AS1) int4*, param 2 an LDS (AS3) int4* (from clang's
// diagnostic on the previous round). Build via integer casts (flat->AS3
// truncates to the 32-bit LDS offset, flat->AS1 keeps the numeric address).
typedef __attribute__((address_space(1))) v4i* g4p_t;
typedef __attribute__((address_space(3))) v4i* l4p_t;
#endif

__device__ __forceinline__ float gelu_exact(float x) {
    return 0.5f * x * (1.0f + erff(x * 0.70710678118654752440f));
}

// One workgroup = one token (b,t). LDS layout (floats):
//   xs[40960] | red[320] | scores[32] | hbuf[128] | wbuf[32]  -> 165,888 bytes
__global__ __launch_bounds__(NTHR, 1)
void TokenWiseLayerPool_fused_kernel(const float* __restrict__ x,
                                     const float* __restrict__ Wp,
                                     const float* __restrict__ bp,
                                     const float* __restrict__ Wg,
                                     const float* __restrict__ bg,
                                     float* __restrict__ out) {
    extern __shared__ float smem[];
    float* xs     = smem;              // 40960
    float* red    = xs + 40960;        // 320
    float* scores = red + 320;         // 32
    float* hbuf   = scores + 32;       // 128
    float* wbuf   = hbuf + 128;        // 32

    const int tid   = threadIdx.x;
    const int lane  = tid & 31;
    const int wave  = tid >> 5;
    const int token = blockIdx.x;      // b*T + t

    // ---- Phase 1: stage x[:, token, :] (L x H = 160 KB) into LDS ----
    // x flat index: l*B*T*H + token*H + h  (layout [L,B,T,H])
    {
        const float* gbase = x + (size_t)token * H_ + (size_t)tid * 4;
        float* lbase = xs + tid * 4;
        #pragma unroll
        for (int l = 0; l < L_; ++l) {
            const float* gp = gbase + (size_t)l * BTH;
            float* lp = lbase + l * H_;
#if HAVE_ASYNC
            __builtin_amdgcn_global_load_async_to_lds_b128(
                (g4p_t)(uintptr_t)gp,
                (l4p_t)(uint32_t)(uintptr_t)lp,
                /*offset=*/0, /*cpol=*/0);
#else
            *(v4f*)lp = *(const v4f*)gp;
#endif
        }
    }
#if HAVE_ASYNC
#if HAVE_WAITASYNC
    __builtin_amdgcn_s_wait_asynccnt(0);
#else
    asm volatile("s_wait_asynccnt 0" ::: "memory");
#endif
#endif
    __syncthreads();

    // ---- Phase 2: per-layer mean over H (scores[l] = mean_h xs[l][h]) ----
    {
        const v4f* xs4 = (const v4f*)xs;
        const int base4 = tid * 32;    // 32 float4 = 128 consecutive floats
        v4f s4 = {0.f, 0.f, 0.f, 0.f};
        #pragma unroll
        for (int i = 0; i < 32; ++i)   // lane-skewed to avoid bank conflicts
            s4 += xs4[base4 + ((i + lane) & 31)];
        red[tid] = s4.x + s4.y + s4.z + s4.w;
    }
    __syncthreads();
    if (tid < L_) {                    // 10 chunks of 128 floats per layer
        float s = 0.f;
        #pragma unroll
        for (int k = 0; k < 10; ++k) s += red[tid * 10 + k];
        scores[tid] = s * (1.0f / H_);
    }
    __syncthreads();

    // ---- Phase 3: h = gelu(scores @ W_proj + b_proj)  (1x32 @ 32x128) ----
#if HAVE_WMMA_F32
    if (tid < 256) {                   // waves 0..7, one 16-wide N tile each
        const int n0   = wave * 16;
        const int half = (lane < 16) ? 0 : 2;
        const int n    = n0 + (lane & 15);
        v8f acc = {0.f, 0.f, 0.f, 0.f, 0.f, 0.f, 0.f, 0.f};
        #pragma unroll
        for (int kk = 0; kk < 8; ++kk) {
            const int k0 = kk * 4;
            // A (16x4 f32): lanes = M rows; only M=0 is live (our token).
            // VGPR0 holds K={0|2}, VGPR1 holds K={1|3} per lane half.
            v2f a; a.x = 0.f; a.y = 0.f;
            if (lane == 0)       { a.x = scores[k0];     a.y = scores[k0 + 1]; }
            else if (lane == 16) { a.x = scores[k0 + 2]; a.y = scores[k0 + 3]; }
            // B (4x16 f32): rows striped across lanes; K split over lane halves.
            v2f b;
            b.x = Wp[(k0 + half)     * HID_ + n];
            b.y = Wp[(k0 + half + 1) * HID_ + n];
            acc = __builtin_amdgcn_wmma_f32_16x16x4_f32(
                false, a, false, b, (short)0, acc, false, false);
        }
        if (lane < 16) {               // D row M=0 lives in VGPR0, lanes 0-15
            hbuf[n0 + lane] = gelu_exact(acc[0] + bp[n0 + lane]);
        }
    }
#else
    if (tid < HID_) {
        float h = bp[tid];
        #pragma unroll
        for (int l = 0; l < L_; ++l) h += scores[l] * Wp[l * HID_ + tid];
        hbuf[tid] = gelu_exact(h);
    }
#endif
    __syncthreads();

    // ---- Phase 4: w = softmax(h @ W_gate + b_gate) over L, by wave 0 ----
    if (tid < 32) {
        float acc = bg[tid];
        #pragma unroll 8
        for (int j = 0; j < HID_; ++j)
            acc += hbuf[j] * Wg[j * L_ + tid];
        float m = acc;
        #pragma unroll
        for (int off = 16; off > 0; off >>= 1)
            m = fmaxf(m, __shfl_xor(m, off, 32));
        float e = expf(acc - m);
        float s = e;
        #pragma unroll
        for (int off = 16; off > 0; off >>= 1)
            s += __shfl_xor(s, off, 32);
        wbuf[tid] = e / s;
    }
    __syncthreads();

    // ---- Phase 5: pooled[h] = sum_l w[l] * xs[l][h]  (float4 per thread) ----
    {
        const v4f* xs4 = (const v4f*)xs;
        v4f acc = {0.f, 0.f, 0.f, 0.f};
        #pragma unroll
        for (int l = 0; l < L_; ++l) {
            const float wl = wbuf[l];
            acc += wl * xs4[l * (H_ / 4) + tid];
        }
        *(v4f*)(out + (size_t)token * H_ + (size_t)tid * 4) = acc;
    }
}

extern "C" void kernel_launch(void* const* d_in, const int* in_sizes, int n_in,
                              void* d_out, int out_size, void* d_ws, size_t ws_size,
                              hipStream_t stream) {
    (void)in_sizes; (void)n_in; (void)out_size; (void)d_ws; (void)ws_size;
    const float* x  = (const float*)d_in[0];
    const float* Wp = (const float*)d_in[1];
    const float* bp = (const float*)d_in[2];
    const float* Wg = (const float*)d_in[3];
    const float* bg = (const float*)d_in[4];
    float* out = (float*)d_out;

    const size_t shmem = (size_t)(40960 + 320 + 32 + 128 + 32) * sizeof(float); // 165,888 B
    // Opt in to >64KB dynamic LDS (CDNA5 WGP has 320KB). Idempotent/deterministic.
    (void)hipFuncSetAttribute((const void*)TokenWiseLayerPool_fused_kernel,
                              hipFuncAttributeMaxDynamicSharedMemorySize, (int)shmem);

    TokenWiseLayerPool_fused_kernel<<<dim3(NTOK), dim3(NTHR), shmem, stream>>>(
        x, Wp, bp, Wg, bg, out);
}